// BandSplitModule_85950885528373
// MI455X (gfx1250) — compile-verified
//
#include <hip/hip_runtime.h>

typedef __attribute__((ext_vector_type(2))) float v2f;
typedef __attribute__((ext_vector_type(8))) float v8f;

#define NBANDS 45
#define NB_B 8
#define NB_C 2
#define NB_F 1025
#define NB_T 512
#define NB_E 128
#define TT 32           // time tile per block
#define XS_STRIDE 40    // LDS row stride (floats); 2*40 mod 64 == 16 -> no half-wave bank clash
#define DMAX 256        // max feature dim (bw=64 -> d=256)
#define LN_EPS 1e-5f

#define AS1 __attribute__((address_space(1)))
#define AS3 __attribute__((address_space(3)))

#if defined(__has_builtin)
#if __has_builtin(__builtin_amdgcn_global_load_async_to_lds_b32)
#define HAVE_ASYNC_LDS 1
#endif
#if __has_builtin(__builtin_amdgcn_s_wait_asynccnt)
#define HAVE_WAIT_ASYNC_BUILTIN 1
#endif
#endif

struct PtrTable { const float* p[1 + 4 * NBANDS]; };

__device__ __forceinline__ void band_spec(int band, int& fs, int& bw) {
    if (band < 32)      { fs = band * 16;              bw = 16; }
    else if (band < 40) { fs = 512 + (band - 32) * 32; bw = 32; }
    else if (band < 44) { fs = 768 + (band - 40) * 64; bw = 64; }
    else                { fs = 1024;                   bw = 1;  }
}

__global__ __launch_bounds__(256) void bandsplit_wmma_kernel(PtrTable tab, float* __restrict__ out) {
    const int ttile = blockIdx.x;
    const int band  = blockIdx.y;
    const int b     = blockIdx.z;
    const int tid   = threadIdx.x;
    const int lane  = tid & 31;
    const int wave  = tid >> 5;

    int fs, bw;
    band_spec(band, fs, bw);
    const int d  = 4 * bw;          // C * reim * bw
    const int t0 = ttile * TT;

    const float* __restrict__ x     = tab.p[0];
    const float* __restrict__ gamma = tab.p[1 + band];
    const float* __restrict__ beta  = tab.p[1 + NBANDS + band];
    const float* __restrict__ W     = tab.p[1 + 2 * NBANDS + band];
    const float* __restrict__ bias  = tab.p[1 + 3 * NBANDS + band];

    __shared__ float xs[DMAX * XS_STRIDE];      // features (later normalized in place), [k][t]
    __shared__ float s_mean[TT], s_rs[TT];
    __shared__ float s_sum[8][TT], s_sum2[8][TT];
    __shared__ float s_g[DMAX], s_b[DMAX];

    // ---- stage band tile into LDS; feature order k = (c*2 + reim)*bw + fb ----
    const int nPairs = NB_C * bw * TT;          // (c, fb, tt)
    for (int idx = tid; idx < nPairs; idx += 256) {
        const int tt   = idx & (TT - 1);
        const int rest = idx >> 5;              // c*bw + fb   (TT == 32)
        const int fb   = rest % bw;
        const int c    = rest / bw;
        const int f    = fs + fb;
        const float* gp = x +
            (((((size_t)b * NB_C + c) * NB_F + f) * NB_T + (size_t)(t0 + tt)) * 2);
        float* l0 = &xs[((c * 2 + 0) * bw + fb) * XS_STRIDE + tt];   // reim = 0
        float* l1 = &xs[((c * 2 + 1) * bw + fb) * XS_STRIDE + tt];   // reim = 1
#ifdef HAVE_ASYNC_LDS
        // CDNA5 async copy: global -> LDS without a VGPR round-trip (ASYNCcnt-tracked)
        __builtin_amdgcn_global_load_async_to_lds_b32((AS1 int*)(gp + 0), (AS3 int*)l0, 0, 0);
        __builtin_amdgcn_global_load_async_to_lds_b32((AS1 int*)(gp + 1), (AS3 int*)l1, 0, 0);
#else
        const float2 v = *(const float2*)gp;
        *l0 = v.x;
        *l1 = v.y;
#endif
    }
    for (int k = tid; k < d; k += 256) { s_g[k] = gamma[k]; s_b[k] = beta[k]; }
#ifdef HAVE_ASYNC_LDS
    // barriers do NOT wait on ASYNCcnt -> drain async LDS writes explicitly
#ifdef HAVE_WAIT_ASYNC_BUILTIN
    __builtin_amdgcn_s_wait_asynccnt(0);
#else
    asm volatile("s_wait_asynccnt 0x0" ::: "memory");
#endif
#endif
    __syncthreads();

    // ---- layernorm statistics: 8 k-chunks x 32 t-lanes ----
    {
        const int t     = tid & (TT - 1);
        const int chunk = tid >> 5;
        float sum = 0.f, sum2 = 0.f;
        for (int k = chunk; k < d; k += 8) {
            const float v = xs[k * XS_STRIDE + t];
            sum += v; sum2 += v * v;
        }
        s_sum[chunk][t] = sum; s_sum2[chunk][t] = sum2;
    }
    __syncthreads();
    if (tid < TT) {
        float sum = 0.f, sum2 = 0.f;
        #pragma unroll
        for (int c = 0; c < 8; ++c) { sum += s_sum[c][tid]; sum2 += s_sum2[c][tid]; }
        const float inv_d = 1.0f / (float)d;
        const float m   = sum * inv_d;
        const float var = sum2 * inv_d - m * m;
        s_mean[tid] = m;
        s_rs[tid]   = rsqrtf(var + LN_EPS);
    }
    __syncthreads();

    // ---- normalize in place ----
    for (int idx = tid; idx < d * TT; idx += 256) {
        const int t = idx & (TT - 1);
        const int k = idx >> 5;
        const float v = xs[k * XS_STRIDE + t];
        xs[k * XS_STRIDE + t] = (v - s_mean[t]) * s_rs[t] * s_g[k] + s_b[k];
    }
    __syncthreads();

    // ---- WMMA f32 16x16x4: each wave owns N-tile `wave`, both M-tiles (t 0-15, 16-31) ----
    const int n0 = wave * 16;
    const int m  = lane & 15;                 // M / N index within tile
    const int kb = (lane >> 4) << 1;          // lanes 0-15 -> K base 0; lanes 16-31 -> K base 2

    v8f acc0 = {}; v8f acc1 = {};

    auto kstep = [&](int k0) {
        const int kk = k0 + kb;
        v2f a0, a1, bf;
        a0.x = xs[(kk    ) * XS_STRIDE + m];        // A[m][kk],   M-tile 0
        a0.y = xs[(kk + 1) * XS_STRIDE + m];
        a1.x = xs[(kk    ) * XS_STRIDE + 16 + m];   // M-tile 1 (t + 16)
        a1.y = xs[(kk + 1) * XS_STRIDE + 16 + m];
        bf.x = W[(size_t)(kk    ) * NB_E + n0 + m]; // B[kk][n] (row-major W, L2-hot)
        bf.y = W[(size_t)(kk + 1) * NB_E + n0 + m];
        acc0 = __builtin_amdgcn_wmma_f32_16x16x4_f32(false, a0, false, bf, (short)0, acc0, false, false);
        acc1 = __builtin_amdgcn_wmma_f32_16x16x4_f32(false, a1, false, bf, (short)0, acc1, false, false);
    };

    int k0 = 0;
    for (; k0 + 8 <= d; k0 += 8) {   // d in {64,128,256}: always even # of K-steps
        kstep(k0);
        kstep(k0 + 4);
    }
    if (k0 < d) kstep(k0);           // tail: only the bw=1 band (d = 4)

    // ---- epilogue: C/D layout -> VGPR j = M row j (+8 for lanes 16-31), N = lane&15 ----
    const float bv  = bias[n0 + m];
    const int  mhi  = (lane >> 4) << 3;
    const size_t rowbase = ((size_t)b * NBANDS + band) * NB_T;
    #pragma unroll
    for (int j = 0; j < 8; ++j) {
        const int ta = t0 + mhi + j;          // M-tile 0
        const int tb = ta + 16;               // M-tile 1
        out[(rowbase + ta) * NB_E + n0 + m] = acc0[j] + bv;
        out[(rowbase + tb) * NB_E + n0 + m] = acc1[j] + bv;
    }
}

extern "C" void kernel_launch(void* const* d_in, const int* in_sizes, int n_in,
                              void* d_out, int out_size, void* d_ws, size_t ws_size,
                              hipStream_t stream) {
    (void)in_sizes; (void)d_ws; (void)ws_size; (void)out_size;
    PtrTable tab;
    const int need = 1 + 4 * NBANDS;
    for (int i = 0; i < need; ++i) tab.p[i] = (i < n_in) ? (const float*)d_in[i] : nullptr;
    dim3 grid(NB_T / TT, NBANDS, NB_B);
    bandsplit_wmma_kernel<<<grid, dim3(256), 0, stream>>>(tab, (float*)d_out);
}